// YOLOXLoss_4552665334090
// MI455X (gfx1250) — compile-verified
//
#include <hip/hip_runtime.h>
#include <cstdint>
#include <cstddef>

// ---------------- problem constants (match reference) ----------------
constexpr int Bsz = 32;          // batch
constexpr int G   = 32;          // GT boxes per image
constexpr int C   = 80;          // classes
constexpr int CH  = 85;          // 4 bbox + 1 obj + 80 cls
constexpr int N   = 8400;        // 6400 + 1600 + 400 anchors
constexpr int BN  = Bsz * N;
constexpr long long BGN = (long long)Bsz * G * N;

typedef float f32x2 __attribute__((ext_vector_type(2)));
typedef float f32x8 __attribute__((ext_vector_type(8)));

// ---------------- helpers ----------------
__device__ __forceinline__ const float* level_base(const float* p0, const float* p1,
                                                   const float* p2, int b, int n,
                                                   int& HW, int& idx) {
  if (n < 6400)      { HW = 6400; idx = n;        return p0 + (size_t)b * CH * 6400; }
  else if (n < 8000) { HW = 1600; idx = n - 6400; return p1 + (size_t)b * CH * 1600; }
  HW = 400; idx = n - 8000;                        return p2 + (size_t)b * CH * 400;
}

__device__ __forceinline__ void anchor_geom(int n, float& cx, float& cy, float& s) {
  if (n < 6400)      { int h = n / 80, w = n % 80;            s = 8.f;  cx = (w + 0.5f) * 8.f;  cy = (h + 0.5f) * 8.f;  }
  else if (n < 8000) { int i = n - 6400; int h = i / 40, w = i % 40; s = 16.f; cx = (w + 0.5f) * 16.f; cy = (h + 0.5f) * 16.f; }
  else               { int i = n - 8000; int h = i / 20, w = i % 20; s = 32.f; cx = (w + 0.5f) * 32.f; cy = (h + 0.5f) * 32.f; }
}

__device__ __forceinline__ float pair_iou(float gx, float gy, float gw, float gh,
                                          float px, float py, float pw, float ph) {
  float tlx = fmaxf(gx - 0.5f * gw, px - 0.5f * pw);
  float tly = fmaxf(gy - 0.5f * gh, py - 0.5f * ph);
  float brx = fminf(gx + 0.5f * gw, px + 0.5f * pw);
  float bry = fminf(gy + 0.5f * gh, py + 0.5f * ph);
  float iw = fmaxf(brx - tlx, 0.f), ih = fmaxf(bry - tly, 0.f);
  float inter = iw * ih;
  return inter / (gw * gh + pw * ph - inter + 1e-16f);
}

__device__ __forceinline__ unsigned long long cost_key(float c, int n) {
  unsigned u = __float_as_uint(c);
  u = (u & 0x80000000u) ? ~u : (u | 0x80000000u);
  return ((unsigned long long)u << 14) | (unsigned)n;   // N < 2^14
}

// ---------------- K0: per-anchor prep (main bandwidth pass) ----------------
__global__ void k0_prep(const float* __restrict__ p0, const float* __restrict__ p1,
                        const float* __restrict__ p2, const float* __restrict__ labels,
                        float4* __restrict__ bboxw, float* __restrict__ obja,
                        float* __restrict__ s0a, float* __restrict__ S0pa,
                        int* __restrict__ fgora, unsigned* __restrict__ valida) {
  int b = blockIdx.y;
  __shared__ float lab[G][5];
  if (threadIdx.x < G) {
    for (int k = 0; k < 5; ++k) lab[threadIdx.x][k] = labels[((size_t)b * G + threadIdx.x) * 5 + k];
  }
  __syncthreads();
  int n = blockIdx.x * blockDim.x + threadIdx.x;
  if (n >= N) return;

  int HW, idx;
  const float* base = level_base(p0, p1, p2, b, n, HW, idx);
  float px = base[(size_t)0 * HW + idx];
  float py = base[(size_t)1 * HW + idx];
  float pw = base[(size_t)2 * HW + idx];
  float ph = base[(size_t)3 * HW + idx];
  float ob = base[(size_t)4 * HW + idx];

  float s0 = 0.f, S0p = 0.f;
  for (int c = 0; c < C; ++c) {
    if ((c & 7) == 0 && c + 8 < C)
      __builtin_prefetch(base + (size_t)(5 + c + 8) * HW + idx, 0, 1);  // global_prefetch_b8
    float cls = base[(size_t)(5 + c) * HW + idx];
    float pj = sqrtf(cls * ob);
    s0  += fmaxf(log1pf(-pj),  -100.f);
    S0p += fmaxf(log1pf(-cls), -100.f);
  }

  float cxa, cya, st;
  anchor_geom(n, cxa, cya, st);
  float r = 2.5f * st;
  unsigned vb = 0u; bool anyb = false, anyc = false;
  for (int g = 0; g < G; ++g) {
    float gx = lab[g][1], gy = lab[g][2], gw = lab[g][3], gh = lab[g][4];
    float db = fminf(fminf(cxa - (gx - 0.5f * gw), cya - (gy - 0.5f * gh)),
                     fminf((gx + 0.5f * gw) - cxa, (gy + 0.5f * gh) - cya));
    float dc = fminf(fminf(cxa - (gx - r), cya - (gy - r)),
                     fminf((gx + r) - cxa, (gy + r) - cya));
    bool ib = db > 0.f, ic = dc > 0.f;
    anyb |= ib; anyc |= ic;
    if (ib && ic) vb |= (1u << g);
  }
  size_t o = (size_t)b * N + n;
  bboxw[o] = make_float4(px, py, pw, ph);
  obja[o] = ob; s0a[o] = s0; S0pa[o] = S0p;
  fgora[o] = (anyb || anyc) ? 1 : 0;
  valida[o] = vb;
}

// ---------------- K1: WMMA one-hot matmul -> cost & iou matrices ----------------
__global__ __launch_bounds__(32)
void k1_cost(const float* __restrict__ p0, const float* __restrict__ p1,
             const float* __restrict__ p2, const float* __restrict__ labels,
             const float4* __restrict__ bboxw, const float* __restrict__ obja,
             const float* __restrict__ s0a, const int* __restrict__ fgora,
             const unsigned* __restrict__ valida,
             float* __restrict__ cost, float* __restrict__ iouw) {
  int tile = blockIdx.x, b = blockIdx.y;
  int n0 = tile * 16, lane = threadIdx.x;

  __shared__ float  A[C][17];        // (l0 - l1)[class][anchor-in-tile]
  __shared__ float4 bb[16];
  __shared__ float  s0l[16], objl[16];
  __shared__ int    fgl[16];
  __shared__ unsigned vl[16];
  __shared__ int    cid[G];
  __shared__ float4 gt[G];

  if (lane < 16) {
    size_t o = (size_t)b * N + n0 + lane;
    bb[lane] = bboxw[o]; s0l[lane] = s0a[o]; objl[lane] = obja[o];
    fgl[lane] = fgora[o]; vl[lane] = valida[o];
  }
  {
    const float* L = labels + ((size_t)b * G + lane) * 5;
    cid[lane] = (int)L[0];
    gt[lane]  = make_float4(L[1], L[2], L[3], L[4]);
  }
  __syncthreads();

  int HW, idx0;
  const float* base = level_base(p0, p1, p2, b, n0, HW, idx0);  // tile never straddles levels
  for (int t = lane; t < C * 16; t += 32) {
    int c = t >> 4, j = t & 15;
    float cls = base[(size_t)(5 + c) * HW + idx0 + j];
    float pj = sqrtf(cls * objl[j]);
    float l1 = fmaxf(logf(pj),    -100.f);
    float l0 = fmaxf(log1pf(-pj), -100.f);
    A[c][j] = l0 - l1;
  }
  __syncthreads();

  // D = OneHot(G x 80) * A(80 x 16) via V_WMMA_F32_16X16X4_F32, K-steps of 4.
  f32x8 clo = {0.f, 0.f, 0.f, 0.f, 0.f, 0.f, 0.f, 0.f};
  f32x8 chi = {0.f, 0.f, 0.f, 0.f, 0.f, 0.f, 0.f, 0.f};
  int m  = lane & 15;
  int kk = (lane < 16) ? 0 : 2;
  for (int kt = 0; kt < C / 4; ++kt) {
    int k0 = kt * 4;
    f32x2 bfrag; bfrag.x = A[k0 + kk][m];     bfrag.y = A[k0 + kk + 1][m];
    f32x2 alo;   alo.x = (cid[m]      == k0 + kk) ? 1.f : 0.f;
                 alo.y = (cid[m]      == k0 + kk + 1) ? 1.f : 0.f;
    f32x2 ahi;   ahi.x = (cid[m + 16] == k0 + kk) ? 1.f : 0.f;
                 ahi.y = (cid[m + 16] == k0 + kk + 1) ? 1.f : 0.f;
    clo = __builtin_amdgcn_wmma_f32_16x16x4_f32(false, alo, false, bfrag, (short)0, clo, false, false);
    chi = __builtin_amdgcn_wmma_f32_16x16x4_f32(false, ahi, false, bfrag, (short)0, chi, false, false);
  }

  // Fuse IoU + penalties, write cost/iou.  D layout: lane<16 -> M=v, N=lane; lane>=16 -> M=v+8, N=lane-16.
  int j = m;
  float4 p = bb[j];
  float pen_fg = fgl[j] ? 0.f : 1e9f;
  for (int half = 0; half < 2; ++half) {
    for (int v = 0; v < 8; ++v) {
      int grow = ((lane < 16) ? v : v + 8) + half * 16;
      float acc = (half == 0) ? clo[v] : chi[v];
      float4 g4 = gt[grow];
      float iv = pair_iou(g4.x, g4.y, g4.z, g4.w, p.x, p.y, p.z, p.w);
      float pen_v = ((vl[j] >> grow) & 1u) ? 0.f : 100000.f;
      float cst = (acc - s0l[j]) - 3.f * logf(iv + 1e-8f) + pen_v + pen_fg;
      size_t o = ((size_t)b * G + grow) * N + n0 + j;
      cost[o] = cst;
      iouw[o] = iv;
    }
  }
}

// ---------------- K2: per (b,g) dyn_k + rank threshold ----------------
__global__ __launch_bounds__(256)
void k2_select(const float* __restrict__ cost, const float* __restrict__ iouw,
               const int* __restrict__ fgora, unsigned long long* __restrict__ thresh) {
  int g = blockIdx.x, b = blockIdx.y, tid = threadIdx.x;
  const int T = 256;
  const float* crow = cost + ((size_t)b * G + g) * N;
  const float* irow = iouw + ((size_t)b * G + g) * N;
  const int*   frow = fgora + (size_t)b * N;

  __shared__ float cf[T * 10];
  __shared__ unsigned long long ck[T * 10];
  __shared__ float rv[T]; __shared__ int ri[T];
  __shared__ unsigned long long rk[T];
  __shared__ float ssum; __shared__ int sdynk;
  __shared__ unsigned long long sth;

  // ---- phase 1: sum of top-10 masked IoU -> dyn_k ----
  float loc[10];
  for (int i = 0; i < 10; ++i) loc[i] = -1.f;         // ascending, loc[0] = min kept
  for (int n = tid; n < N; n += T) {
    float v = frow[n] ? irow[n] : 0.f;
    if (v > loc[0]) {
      int i = 1;
      while (i < 10 && v > loc[i]) { loc[i - 1] = loc[i]; ++i; }
      loc[i - 1] = v;
    }
  }
  for (int i = 0; i < 10; ++i) cf[tid * 10 + i] = loc[i];
  if (tid == 0) ssum = 0.f;
  __syncthreads();
  for (int r = 0; r < 10; ++r) {
    float mv = -2.f; int mi = tid * 10;
    for (int i = 0; i < 10; ++i)
      if (cf[tid * 10 + i] > mv) { mv = cf[tid * 10 + i]; mi = tid * 10 + i; }
    rv[tid] = mv; ri[tid] = mi;
    __syncthreads();
    for (int s = T / 2; s > 0; s >>= 1) {
      if (tid < s && rv[tid + s] > rv[tid]) { rv[tid] = rv[tid + s]; ri[tid] = ri[tid + s]; }
      __syncthreads();
    }
    if (tid == 0) { ssum += fmaxf(rv[0], 0.f); cf[ri[0]] = -2.f; }
    __syncthreads();
  }
  if (tid == 0) sdynk = max((int)ssum, 1);
  __syncthreads();
  int K = sdynk;

  // ---- phase 2: dyn_k-th smallest (cost,idx) key ----
  unsigned long long lk[10];
  for (int i = 0; i < 10; ++i) lk[i] = ~0ull;         // descending, lk[0] = max kept
  for (int n = tid; n < N; n += T) {
    unsigned long long key = cost_key(crow[n], n);
    if (key < lk[0]) {
      int i = 1;
      while (i < 10 && key < lk[i]) { lk[i - 1] = lk[i]; ++i; }
      lk[i - 1] = key;
    }
  }
  for (int i = 0; i < 10; ++i) ck[tid * 10 + i] = lk[i];
  __syncthreads();
  for (int r = 0; r < K; ++r) {
    unsigned long long mv = ~0ull; int mi = tid * 10;
    for (int i = 0; i < 10; ++i)
      if (ck[tid * 10 + i] < mv) { mv = ck[tid * 10 + i]; mi = tid * 10 + i; }
    rk[tid] = mv; ri[tid] = mi;
    __syncthreads();
    for (int s = T / 2; s > 0; s >>= 1) {
      if (tid < s && rk[tid + s] < rk[tid]) { rk[tid] = rk[tid + s]; ri[tid] = ri[tid + s]; }
      __syncthreads();
    }
    if (tid == 0) { sth = rk[0]; ck[ri[0]] = ~0ull; }
    __syncthreads();
  }
  if (tid == 0) thresh[b * G + g] = sth;
}

// ---------------- K3: resolve matches, fused loss partials ----------------
__global__ __launch_bounds__(256)
void k3_loss(const float* __restrict__ p0, const float* __restrict__ p1,
             const float* __restrict__ p2, const float* __restrict__ labels,
             const float* __restrict__ cost, const float* __restrict__ iouw,
             const float4* __restrict__ bboxw, const float* __restrict__ obja,
             const float* __restrict__ S0pa,
             const unsigned long long* __restrict__ thresh,
             float* __restrict__ part) {
  int b = blockIdx.y, tid = threadIdx.x;
  __shared__ int    cidl[G];
  __shared__ float4 gtl[G];
  __shared__ unsigned long long thl[G];
  if (tid < G) {
    const float* L = labels + ((size_t)b * G + tid) * 5;
    cidl[tid] = (int)L[0];
    gtl[tid]  = make_float4(L[1], L[2], L[3], L[4]);
    thl[tid]  = thresh[b * G + tid];
  }
  __syncthreads();

  int n = blockIdx.x * blockDim.x + tid;
  float a_obj = 0.f, a_iou = 0.f, a_cls = 0.f, a_fg = 0.f;
  if (n < N) {
    unsigned mbits = 0u; float bestc = 3.4e38f; int amin = 0;
    for (int g = 0; g < G; ++g) {
      float cst = cost[((size_t)b * G + g) * N + n];
      if (cost_key(cst, n) <= thl[g]) mbits |= (1u << g);
      if (cst < bestc) { bestc = cst; amin = g; }
    }
    if (__popc(mbits) > 1) mbits = (1u << amin);
    bool fg = (mbits != 0u);

    float ob = obja[(size_t)b * N + n];
    float l1o = fmaxf(logf(ob), -100.f), l0o = fmaxf(log1pf(-ob), -100.f);
    a_obj = fg ? -l1o : -l0o;

    if (fg) {
      int mgt = __ffs((int)mbits) - 1;
      float miou = iouw[((size_t)b * G + mgt) * N + n];
      float4 g4 = gtl[mgt];
      float4 pb = bboxw[(size_t)b * N + n];
      float iv = pair_iou(g4.x, g4.y, g4.z, g4.w, pb.x, pb.y, pb.z, pb.w);
      a_iou = 1.f - iv * iv;
      int HW, idx;
      const float* base = level_base(p0, p1, p2, b, n, HW, idx);
      float clsp = base[(size_t)(5 + cidl[mgt]) * HW + idx];
      float l1c = fmaxf(logf(clsp), -100.f), l0c = fmaxf(log1pf(-clsp), -100.f);
      a_cls = -miou * l1c - S0pa[(size_t)b * N + n] + miou * l0c;
      a_fg = 1.f;
    }
  }

  __shared__ float red[4][256];
  red[0][tid] = a_obj; red[1][tid] = a_iou; red[2][tid] = a_cls; red[3][tid] = a_fg;
  __syncthreads();
  for (int s = 128; s > 0; s >>= 1) {
    if (tid < s)
      for (int k = 0; k < 4; ++k) red[k][tid] += red[k][tid + s];
    __syncthreads();
  }
  if (tid == 0) {
    int pblk = blockIdx.y * gridDim.x + blockIdx.x;
    for (int k = 0; k < 4; ++k) part[(size_t)pblk * 4 + k] = red[k][0];
  }
}

// ---------------- K4: deterministic final reduce ----------------
__global__ __launch_bounds__(256)
void k4_final(const float* __restrict__ part, int nblk, float* __restrict__ out) {
  int tid = threadIdx.x;
  __shared__ float red[4][256];
  float a[4] = {0.f, 0.f, 0.f, 0.f};
  for (int i = tid; i < nblk; i += 256)
    for (int k = 0; k < 4; ++k) a[k] += part[(size_t)i * 4 + k];
  for (int k = 0; k < 4; ++k) red[k][tid] = a[k];
  __syncthreads();
  for (int s = 128; s > 0; s >>= 1) {
    if (tid < s)
      for (int k = 0; k < 4; ++k) red[k][tid] += red[k][tid + s];
    __syncthreads();
  }
  if (tid == 0) {
    float s_obj = red[0][0], s_iou = red[1][0], s_cls = red[2][0], s_fg = red[3][0];
    float gt_num = fmaxf(s_fg, 1.f);
    float li = 5.f * s_iou / gt_num;
    float lo = s_obj / gt_num;
    float lc = s_cls / gt_num;
    out[0] = li + lo + lc; out[1] = li; out[2] = lo; out[3] = lc; out[4] = gt_num;
  }
}

// ---------------- host launcher ----------------
extern "C" void kernel_launch(void* const* d_in, const int* in_sizes, int n_in,
                              void* d_out, int out_size, void* d_ws, size_t ws_size,
                              hipStream_t stream) {
  const float* p0     = (const float*)d_in[0];
  const float* p1     = (const float*)d_in[1];
  const float* p2     = (const float*)d_in[2];
  const float* labels = (const float*)d_in[3];
  float* out = (float*)d_out;

  // workspace carve-up
  float* ws   = (float*)d_ws;
  float* cost = ws;
  float* iouw = cost + (size_t)BGN;
  float4* bboxw = (float4*)(iouw + (size_t)BGN);
  float* obja = (float*)(bboxw + (size_t)BN);
  float* s0a  = obja + (size_t)BN;
  float* S0pa = s0a + (size_t)BN;
  int*   fgora  = (int*)(S0pa + (size_t)BN);
  unsigned* valida = (unsigned*)(fgora + (size_t)BN);
  unsigned long long* thresh = (unsigned long long*)(valida + (size_t)BN);
  float* part = (float*)(thresh + Bsz * G);

  dim3 g0((N + 255) / 256, Bsz);
  k0_prep<<<g0, 256, 0, stream>>>(p0, p1, p2, labels, bboxw, obja, s0a, S0pa, fgora, valida);

  dim3 g1(N / 16, Bsz);
  k1_cost<<<g1, 32, 0, stream>>>(p0, p1, p2, labels, bboxw, obja, s0a, fgora, valida, cost, iouw);

  dim3 g2(G, Bsz);
  k2_select<<<g2, 256, 0, stream>>>(cost, iouw, fgora, thresh);

  dim3 g3((N + 255) / 256, Bsz);
  k3_loss<<<g3, 256, 0, stream>>>(p0, p1, p2, labels, cost, iouw, bboxw, obja, S0pa, thresh, part);

  int nblk = g3.x * g3.y;
  k4_final<<<1, 256, 0, stream>>>(part, nblk, out);
}